// DirGNN_62208306315974
// MI455X (gfx1250) — compile-verified
//
#include <hip/hip_runtime.h>
#include <hip/hip_bf16.h>

// ---------------------------------------------------------------------------
// DirGNN (3x DirGNNConv + global mean pool) for MI455X / gfx1250.
// GEMMs use V_WMMA_F32_16X16X4_F32 (exact fp32 matrix path, wave32).
// Edge aggregation is L2-resident gather + f32 atomic scatter-add.
// ---------------------------------------------------------------------------

typedef __attribute__((ext_vector_type(2))) float v2f;
typedef __attribute__((ext_vector_type(8))) float v8f;

#define HIDP 128   // padded feature width (HID=128, OUT_C=120 padded to 128)

// ---- weight pad/copy: W [K x Co] row-major -> Wp [K x 128], zero padded ----
__global__ void k_pad_weight(const float* __restrict__ W, float* __restrict__ Wp,
                             int K, int Co) {
  int i = blockIdx.x * blockDim.x + threadIdx.x;
  if (i >= K * HIDP) return;
  int r = i >> 7, c = i & (HIDP - 1);
  Wp[i] = (c < Co) ? W[r * Co + c] : 0.0f;
}

// ---- degree histograms (float counts via atomics) ----
__global__ void k_degree(const int* __restrict__ src, const int* __restrict__ dst,
                         float* __restrict__ deg_d, float* __restrict__ deg_s, int E) {
  int e = blockIdx.x * blockDim.x + threadIdx.x;
  if (e >= E) return;
  atomicAdd(&deg_d[dst[e]], 1.0f);
  atomicAdd(&deg_s[src[e]], 1.0f);
}

// ---- deg -> deg^{-1/2} in place (0 where deg==0) ----
__global__ void k_dinv(float* __restrict__ deg, int n) {
  int i = blockIdx.x * blockDim.x + threadIdx.x;
  if (i >= n) return;
  float d = deg[i];
  deg[i] = (d > 0.0f) ? rsqrtf(d) : 0.0f;
}

// ---- per-edge weights, 0.5 (= ALPHA = 1-ALPHA) folded in ----
__global__ void k_edgew(const int* __restrict__ src, const int* __restrict__ dst,
                        const float* __restrict__ dinv_d, const float* __restrict__ dinv_s,
                        float* __restrict__ w2f, float* __restrict__ w2r, int E) {
  int e = blockIdx.x * blockDim.x + threadIdx.x;
  if (e >= E) return;
  int s = src[e], d = dst[e];
  w2f[e] = 0.5f * dinv_d[s] * dinv_d[d];   // forward conv norm
  w2r[e] = 0.5f * dinv_s[s] * dinv_s[d];   // reversed conv norm
}

// ---- WMMA fp32 GEMM: H[N x 128] = X[N x K](ld=ldx) @ Wp[K x 128] ----------
// One wave per 16x16 tile; block = 8 waves = one row tile x 8 col tiles.
// A-matrix f32 16x4 layout: lanes 0-15 hold M=0..15 with {K=0,K=1} in the two
// VGPRs, lanes 16-31 hold {K=2,K=3}. B 4x16: lanes 0-15 = K rows 0/1 cols,
// lanes 16-31 = K rows 2/3. C/D: VGPR r -> row r (lanes 0-15) / r+8 (16-31).
__global__ __launch_bounds__(256)
void k_gemm_wmma(const float* __restrict__ X, int ldx, int K,
                 const float* __restrict__ Wp, float* __restrict__ H) {
  const int lane = threadIdx.x & 31;
  const int wave = threadIdx.x >> 5;
  const int tM   = blockIdx.x * 16;        // row tile
  const int tN   = wave * 16;              // col tile (8 per block = 128 cols)
  const int r    = lane & 15;
  const int koff = (lane >> 4) << 1;       // 0 for lanes 0-15, 2 for 16-31

  v8f c = {};
  const float* xrow = X + (long)(tM + r) * ldx;
  for (int k = 0; k < K; k += 4) {
    v2f a, b;
    a.x = xrow[k + koff];
    a.y = xrow[k + koff + 1];
    b.x = Wp[(k + koff) * HIDP + tN + r];
    b.y = Wp[(k + koff + 1) * HIDP + tN + r];
    c = __builtin_amdgcn_wmma_f32_16x16x4_f32(
        /*neg_a=*/false, a, /*neg_b=*/false, b,
        /*c_mod=*/(short)0, c, /*reuse_a=*/false, /*reuse_b=*/false);
  }
  const int rbase = tM + ((lane >> 4) << 3);
  float* hp = H + (long)rbase * HIDP + tN + r;
#pragma unroll
  for (int i = 0; i < 8; ++i) hp[(long)i * HIDP] = c[i];
}

// scalar GEMM fallback for row-tail (N % 16 != 0; unused for N=100000)
__global__ void k_gemm_tail(const float* __restrict__ X, int ldx, int K,
                            const float* __restrict__ Wp, float* __restrict__ H,
                            int row0, int rows) {
  int idx = blockIdx.x * blockDim.x + threadIdx.x;
  if (idx >= rows * HIDP) return;
  int r = row0 + (idx >> 7), c = idx & (HIDP - 1);
  float acc = 0.0f;
  for (int k = 0; k < K; ++k) acc += X[(long)r * ldx + k] * Wp[k * HIDP + c];
  H[(long)r * HIDP + c] = acc;
}

// ---- fused bidirectional edge aggregation ---------------------------------
// thread = (edge, 4-feature chunk). acc[dst] += w2f*H[src]; acc[src] += w2r*Hr[dst]
__global__ void k_aggregate(const int* __restrict__ src, const int* __restrict__ dst,
                            const float* __restrict__ H, const float* __restrict__ Hr,
                            const float* __restrict__ w2f, const float* __restrict__ w2r,
                            float* __restrict__ acc, long E) {
  long idx = (long)blockIdx.x * blockDim.x + threadIdx.x;
  if (idx >= E * (HIDP / 4)) return;
  long e = idx >> 5;                       // HIDP/4 == 32 chunks per edge
  int  c = (int)(idx & 31) << 2;
  int  s = src[e], d = dst[e];
  float wf = w2f[e], wr = w2r[e];
  const float4 hv = *(const float4*)(H  + (long)s * HIDP + c);
  const float4 rv = *(const float4*)(Hr + (long)d * HIDP + c);
  float* ad = acc + (long)d * HIDP + c;
  float* as = acc + (long)s * HIDP + c;
  atomicAdd(ad + 0, hv.x * wf); atomicAdd(ad + 1, hv.y * wf);
  atomicAdd(ad + 2, hv.z * wf); atomicAdd(ad + 3, hv.w * wf);
  atomicAdd(as + 0, rv.x * wr); atomicAdd(as + 1, rv.y * wr);
  atomicAdd(as + 2, rv.z * wr); atomicAdd(as + 3, rv.w * wr);
}

// ---- combine: acc += 0.5*(b + b_r), optional ReLU, in place ----------------
__global__ void k_combine(float* __restrict__ acc, const float* __restrict__ b,
                          const float* __restrict__ br, int Co, int relu, long n) {
  long i = (long)blockIdx.x * blockDim.x + threadIdx.x;
  if (i >= n) return;
  int c = (int)(i & (HIDP - 1));
  float bias = (c < Co) ? 0.5f * (b[c] + br[c]) : 0.0f;
  float v = acc[i] + bias;
  if (relu) v = fmaxf(v, 0.0f);
  acc[i] = v;
}

// ---- global mean pool -----------------------------------------------------
__global__ void k_count(const int* __restrict__ seg, float* __restrict__ cnt, int N) {
  int i = blockIdx.x * blockDim.x + threadIdx.x;
  if (i >= N) return;
  atomicAdd(&cnt[seg[i]], 1.0f);
}

__global__ void k_poolsum(const float* __restrict__ A, const int* __restrict__ seg,
                          float* __restrict__ out, int N, int Co) {
  long idx = (long)blockIdx.x * blockDim.x + threadIdx.x;
  if (idx >= (long)N * Co) return;
  int n = (int)(idx / Co);
  int c = (int)(idx - (long)n * Co);
  atomicAdd(&out[(long)seg[n] * Co + c], A[(long)n * HIDP + c]);
}

__global__ void k_pooldiv(float* __restrict__ out, const float* __restrict__ cnt,
                          int Co, int total) {
  int i = blockIdx.x * blockDim.x + threadIdx.x;
  if (i >= total) return;
  out[i] = out[i] / fmaxf(cnt[i / Co], 1.0f);
}

// ---------------------------------------------------------------------------
extern "C" void kernel_launch(void* const* d_in, const int* in_sizes, int n_in,
                              void* d_out, int out_size, void* d_ws, size_t ws_size,
                              hipStream_t stream) {
  const float* x       = (const float*)d_in[0];
  const int*   eidx    = (const int*)d_in[1];
  const int*   seg     = (const int*)d_in[2];
  const float* W_in    = (const float*)d_in[3];
  const float* b_in    = (const float*)d_in[4];
  const float* W_in_r  = (const float*)d_in[5];
  const float* b_in_r  = (const float*)d_in[6];
  const float* W_mid   = (const float*)d_in[7];
  const float* b_mid   = (const float*)d_in[8];
  const float* W_mid_r = (const float*)d_in[9];
  const float* b_mid_r = (const float*)d_in[10];
  const float* W_out   = (const float*)d_in[11];
  const float* b_out   = (const float*)d_in[12];
  const float* W_out_r = (const float*)d_in[13];
  const float* b_out_r = (const float*)d_in[14];
  float* out = (float*)d_out;

  const int N     = in_sizes[2];          // 100000 (batch_seg length)
  const int E     = in_sizes[1] / 2;      // 1600000
  const int IN_C  = in_sizes[0] / N;      // 4
  const int OUT_C = in_sizes[12];         // 120 (len(b_out))
  const int G     = out_size / OUT_C;     // 256
  const int* src = eidx;
  const int* dst = eidx + E;

  // ---- workspace layout (floats, 16B-aligned sections) ----
  float* ws = (float*)d_ws;
  auto al4 = [](size_t v) { return (v + 3) & ~(size_t)3; };
  size_t off = 0;
  float* Wp[6];
  for (int i = 0; i < 6; ++i) { Wp[i] = ws + off; off += (size_t)HIDP * HIDP; }
  float* dinv_d = ws + off; off += al4((size_t)N);
  float* dinv_s = ws + off; off += al4((size_t)N);
  float* w2f    = ws + off; off += al4((size_t)E);
  float* w2r    = ws + off; off += al4((size_t)E);
  float* H      = ws + off; off += (size_t)N * HIDP;
  float* Hr     = ws + off; off += (size_t)N * HIDP;
  float* A0     = ws + off; off += (size_t)N * HIDP;
  float* A1     = ws + off; off += (size_t)N * HIDP;
  float* cnt    = ws + off; off += al4((size_t)G);
  (void)ws_size; (void)n_in;

  // ---- pad weights to uniform [K x 128] ----
  const float* Wsrc[6] = {W_in, W_in_r, W_mid, W_mid_r, W_out, W_out_r};
  const int    Ks[6]   = {IN_C, IN_C, HIDP, HIDP, HIDP, HIDP};
  const int    Cos[6]  = {HIDP, HIDP, HIDP, HIDP, OUT_C, OUT_C};
  for (int i = 0; i < 6; ++i) {
    int n = Ks[i] * HIDP;
    k_pad_weight<<<(n + 255) / 256, 256, 0, stream>>>(Wsrc[i], Wp[i], Ks[i], Cos[i]);
  }

  // ---- degrees -> dinv -> per-edge weights (layer-invariant) ----
  hipMemsetAsync(dinv_d, 0, sizeof(float) * al4((size_t)N) * 2, stream);
  k_degree<<<(E + 255) / 256, 256, 0, stream>>>(src, dst, dinv_d, dinv_s, E);
  k_dinv<<<(N + 255) / 256, 256, 0, stream>>>(dinv_d, N);
  k_dinv<<<(N + 255) / 256, 256, 0, stream>>>(dinv_s, N);
  k_edgew<<<(E + 255) / 256, 256, 0, stream>>>(src, dst, dinv_d, dinv_s, w2f, w2r, E);

  // ---- one DirGNNConv layer ----
  const int rowTiles = N / 16;
  const int tailRows = N % 16;
  auto layer = [&](const float* X, int ldx, int K, const float* WA, const float* WB,
                   const float* b, const float* br, int Co, int relu, float* ACC) {
    k_gemm_wmma<<<rowTiles, 256, 0, stream>>>(X, ldx, K, WA, H);
    k_gemm_wmma<<<rowTiles, 256, 0, stream>>>(X, ldx, K, WB, Hr);
    if (tailRows) {
      int nt = tailRows * HIDP;
      k_gemm_tail<<<(nt + 255) / 256, 256, 0, stream>>>(X, ldx, K, WA, H, rowTiles * 16, tailRows);
      k_gemm_tail<<<(nt + 255) / 256, 256, 0, stream>>>(X, ldx, K, WB, Hr, rowTiles * 16, tailRows);
    }
    hipMemsetAsync(ACC, 0, sizeof(float) * (size_t)N * HIDP, stream);
    long tot = (long)E * (HIDP / 4);
    k_aggregate<<<(unsigned)((tot + 255) / 256), 256, 0, stream>>>(src, dst, H, Hr, w2f, w2r, ACC, E);
    long n2 = (long)N * HIDP;
    k_combine<<<(unsigned)((n2 + 255) / 256), 256, 0, stream>>>(ACC, b, br, Co, relu, n2);
  };

  layer(x,  IN_C, IN_C, Wp[0], Wp[1], b_in,  b_in_r,  HIDP,  1, A0);  // layer 1 + ReLU
  layer(A0, HIDP, HIDP, Wp[2], Wp[3], b_mid, b_mid_r, HIDP,  1, A1);  // layer 2 + ReLU
  layer(A1, HIDP, HIDP, Wp[4], Wp[5], b_out, b_out_r, OUT_C, 0, A0);  // layer 3

  // ---- global mean pool ----
  hipMemsetAsync(out, 0, sizeof(float) * (size_t)out_size, stream);
  hipMemsetAsync(cnt, 0, sizeof(float) * al4((size_t)G), stream);
  k_count<<<(N + 255) / 256, 256, 0, stream>>>(seg, cnt, N);
  long np = (long)N * OUT_C;
  k_poolsum<<<(unsigned)((np + 255) / 256), 256, 0, stream>>>(A0, seg, out, N, OUT_C);
  int gp = G * OUT_C;
  k_pooldiv<<<(gp + 255) / 256, 256, 0, stream>>>(out, cnt, OUT_C, gp);
}